// SAGPool_15195594293503
// MI455X (gfx1250) — compile-verified
//
#include <hip/hip_runtime.h>
#include <hip/hip_bf16.h>
#include <math.h>

typedef __attribute__((ext_vector_type(2))) float v2f;
typedef __attribute__((ext_vector_type(8))) float v8f;

#define D_DIM 256
#define NPER  512
#define KEEPN 256   // ceil(0.5 * NPER)

// ---------------- zero workspace ----------------
__global__ void zero_kernel(unsigned* __restrict__ deg_out, unsigned* __restrict__ deg_in,
                            float* __restrict__ agg, unsigned* __restrict__ keep, int N) {
    int i = blockIdx.x * blockDim.x + threadIdx.x;
    if (i < N) { deg_out[i] = 0u; deg_in[i] = 0u; agg[i] = 0.0f; keep[i] = 0u; }
}

// ---------------- edge-parallel degree counts ----------------
__global__ void degree_kernel(const int* __restrict__ src, const int* __restrict__ dst,
                              unsigned* __restrict__ deg_out, unsigned* __restrict__ deg_in, int E) {
    int e = blockIdx.x * blockDim.x + threadIdx.x;
    if (e < E) {
        atomicAdd(&deg_out[src[e]], 1u);
        atomicAdd(&deg_in[dst[e]], 1u);
    }
}

// ---------------- h = x @ W via V_WMMA_F32_16X16X4_F32, then out-degree norm ----------------
// One wave handles a 16-row tile of x. A = x[row0:row0+16, k0:k0+4] (f32 16x4),
// B = W[k0:k0+4] broadcast across all 16 columns (4x16). Every column of D equals h;
// column 0 lives in lanes 0 (M=0..7 -> acc[0..7]) and 16 (M=8..15).
__global__ void gemv_wmma_kernel(const float* __restrict__ x, const float* __restrict__ W,
                                 const unsigned* __restrict__ deg_out,
                                 float* __restrict__ hsc, int N) {
    __shared__ float Wsh[D_DIM];
    int tid = threadIdx.x;
    Wsh[tid] = W[tid];          // blockDim.x == D_DIM == 256
    __syncthreads();

    int wave = blockIdx.x * (blockDim.x >> 5) + (tid >> 5);
    int lane = tid & 31;
    int row0 = wave * 16;
    if (row0 >= N) return;      // uniform per wave; EXEC stays all-1s for live waves

    int m = lane & 15;                 // A-matrix row within tile (lanes 0-15 / 16-31)
    int khalf = (lane >> 4) << 1;      // 0 for lanes 0-15 (K=0,1), 2 for lanes 16-31 (K=2,3)
    const float* xr = x + (size_t)(row0 + m) * D_DIM + khalf;

    v8f acc = {};
    #pragma unroll 8
    for (int k0 = 0; k0 < D_DIM; k0 += 4) {
        v2f a;  a.x  = xr[k0];             a.y  = xr[k0 + 1];
        v2f bb; bb.x = Wsh[k0 + khalf];    bb.y = Wsh[k0 + khalf + 1];
        acc = __builtin_amdgcn_wmma_f32_16x16x4_f32(
            /*neg_a=*/false, a, /*neg_b=*/false, bb,
            /*c_mod=*/(short)0, acc, /*reuse_a=*/false, /*reuse_b=*/false);
    }

    if (m == 0) {   // lanes 0 and 16 hold column N=0 of D
        int rbase = row0 + (lane >> 4) * 8;
        #pragma unroll
        for (int v = 0; v < 8; ++v) {
            float h  = acc[v];
            float dg = (float)deg_out[rbase + v];
            hsc[rbase + v] = h * rsqrtf(fmaxf(dg, 1.0f));
        }
    }
}

// ---------------- agg[dst] += hsc[src] ----------------
__global__ void aggregate_kernel(const int* __restrict__ src, const int* __restrict__ dst,
                                 const float* __restrict__ hsc, float* __restrict__ agg, int E) {
    int e = blockIdx.x * blockDim.x + threadIdx.x;
    if (e < E) atomicAdd(&agg[dst[e]], hsc[src[e]]);
}

// ---------------- per-graph score + bitonic top-k ----------------
// Total order: score descending, index ascending on ties (matches lax.top_k stability).
__global__ void topk_kernel(const float* __restrict__ agg, const unsigned* __restrict__ deg_in,
                            const float* __restrict__ bptr,
                            int* __restrict__ perm_i, float* __restrict__ ptanh,
                            unsigned* __restrict__ keep,
                            float* __restrict__ out_perm, float* __restrict__ out_nnn) {
    __shared__ float svals[NPER];
    __shared__ int   sidx[NPER];
    int g = blockIdx.x, tid = threadIdx.x;
    float bval = bptr[0];

    for (int j = tid; j < NPER; j += blockDim.x) {
        int node = g * NPER + j;
        float s = agg[node] * rsqrtf(fmaxf((float)deg_in[node], 1.0f)) + bval;
        svals[j] = s;
        sidx[j]  = j;
    }

    for (int k = 2; k <= NPER; k <<= 1) {
        for (int j = k >> 1; j > 0; j >>= 1) {
            __syncthreads();
            for (int t = tid; t < NPER; t += blockDim.x) {
                int ixj = t ^ j;
                if (ixj > t) {
                    float sa = svals[t],  sb = svals[ixj];
                    int   ia = sidx[t],   ib = sidx[ixj];
                    bool inOrder = (sa > sb) || (sa == sb && ia < ib);
                    bool up = ((t & k) == 0);
                    if (up != inOrder) {
                        svals[t]   = sb; sidx[t]   = ib;
                        svals[ixj] = sa; sidx[ixj] = ia;
                    }
                }
            }
        }
    }
    __syncthreads();

    for (int j = tid; j < KEEPN; j += blockDim.x) {
        int node = g * NPER + sidx[j];
        int r    = g * KEEPN + j;
        perm_i[r]  = node;
        ptanh[r]   = tanhf(svals[j]);
        keep[node] = 1u;
        out_perm[r] = (float)node;        // node ids < 2^24: exact in f32
    }
    if (tid == 0) out_nnn[g] = (float)KEEPN;
}

// ---------------- feat = x[perm] * tanh(score[perm]), float4 coalesced ----------------
__global__ void feat_kernel(const float* __restrict__ x, const int* __restrict__ perm_i,
                            const float* __restrict__ ptanh, float* __restrict__ out_feat, int R) {
    int t   = blockIdx.x * blockDim.x + threadIdx.x;
    int row = t >> 6;              // 64 float4 per 256-wide row
    int c4  = t & 63;
    if (row < R) {
        int   node = perm_i[row];
        float s    = ptanh[row];
        float4 v = ((const float4*)(x + (size_t)node * D_DIM))[c4];
        v.x *= s; v.y *= s; v.z *= s; v.w *= s;
        ((float4*)(out_feat + (size_t)row * D_DIM))[c4] = v;
    }
}

// ---------------- edge mask ----------------
__global__ void mask_kernel(const int* __restrict__ src, const int* __restrict__ dst,
                            const unsigned* __restrict__ keep, float* __restrict__ out_mask, int E) {
    int e = blockIdx.x * blockDim.x + threadIdx.x;
    if (e < E) out_mask[e] = (keep[src[e]] && keep[dst[e]]) ? 1.0f : 0.0f;
}

extern "C" void kernel_launch(void* const* d_in, const int* in_sizes, int n_in,
                              void* d_out, int out_size, void* d_ws, size_t ws_size,
                              hipStream_t stream) {
    const float* x   = (const float*)d_in[0];
    const float* W   = (const float*)d_in[1];
    const float* b   = (const float*)d_in[2];
    const int*   src = (const int*)d_in[3];
    const int*   dst = (const int*)d_in[4];

    int N = in_sizes[0] / D_DIM;   // 131072
    int E = in_sizes[3];           // 2097152
    int B = in_sizes[5];           // 256
    int R = B * KEEPN;             // 65536 kept rows

    char* ws = (char*)d_ws;
    unsigned* deg_out = (unsigned*)ws;  ws += (size_t)N * 4;
    unsigned* deg_in  = (unsigned*)ws;  ws += (size_t)N * 4;
    float*    agg     = (float*)ws;     ws += (size_t)N * 4;
    float*    hsc     = (float*)ws;     ws += (size_t)N * 4;
    unsigned* keep    = (unsigned*)ws;  ws += (size_t)N * 4;
    int*      perm_i  = (int*)ws;       ws += (size_t)R * 4;
    float*    ptanh   = (float*)ws;     ws += (size_t)R * 4;

    float* out_feat = (float*)d_out;                       // [R, 256]
    float* out_perm = out_feat + (size_t)R * D_DIM;        // [R]
    float* out_mask = out_perm + R;                        // [E]
    float* out_nnn  = out_mask + E;                        // [B]

    const int thr = 256;
    zero_kernel<<<(N + thr - 1) / thr, thr, 0, stream>>>(deg_out, deg_in, agg, keep, N);
    degree_kernel<<<(E + thr - 1) / thr, thr, 0, stream>>>(src, dst, deg_out, deg_in, E);
    int nwaves = N / 16;                                   // 16 rows per wave
    gemv_wmma_kernel<<<nwaves / 8, thr, 0, stream>>>(x, W, deg_out, hsc, N);
    aggregate_kernel<<<(E + thr - 1) / thr, thr, 0, stream>>>(src, dst, hsc, agg, E);
    topk_kernel<<<B, thr, 0, stream>>>(agg, deg_in, b, perm_i, ptanh, keep, out_perm, out_nnn);
    feat_kernel<<<((size_t)R * 64 + thr - 1) / thr, thr, 0, stream>>>(x, perm_i, ptanh, out_feat, R);
    mask_kernel<<<(E + thr - 1) / thr, thr, 0, stream>>>(src, dst, keep, out_mask, E);
}